// QueryAndGroup_47373489274982
// MI455X (gfx1250) — compile-verified
//
#include <hip/hip_runtime.h>

typedef __attribute__((ext_vector_type(2))) float v2f;
typedef __attribute__((ext_vector_type(8))) float v8f;

#define NB      8
#define NPTS    8192
#define NQ      2048
#define NSAMP   64
#define NCH     64
#define OCH     67      // 3 + 64
#define R2      0.04f   // 0.2^2

// One wave (32 threads) handles a tile of 16 query points:
//  - ball query via V_WMMA_F32_16X16X4_F32 (D = d^2 - r^2 for a 16x16 tile)
//  - ordered compaction of in-ball indices (ascending index == reference top-k),
//    with per-row sample counts kept in registers (uniform per half-wave)
//  - fused gather + subtract + concat write of the (16, 64, 67) output block
__global__ __launch_bounds__(32) void ballquery_group_wmma(
    const float* __restrict__ xyz,      // (B, NPTS, 3)
    const float* __restrict__ nxyz,     // (B, NQ, 3)
    const float* __restrict__ points,   // (B, NPTS, NCH)
    float* __restrict__ out)            // (B, NQ, NSAMP, OCH)
{
    const int lane = threadIdx.x;           // 0..31
    const int b    = blockIdx.x >> 7;       // 128 tiles per batch
    const int m0   = (blockIdx.x & 127) << 4;

    __shared__ float qn2[16];               // |q|^2 - r^2 per row
    __shared__ float qc[16][3];             // query coords per row
    __shared__ int   cnt[16];               // samples found per row (post-scan)
    __shared__ int   ibuf[16][NSAMP];       // collected indices per row

    const int m = lane & 15;

    // ---- load query tile (lanes 16..31 load duplicates of rows 0..15) ----
    const float* qp = nxyz + ((size_t)b * NQ + m0 + m) * 3;
    float qx = qp[0], qy = qp[1], qz = qp[2];
    if (lane < 16) {
        qn2[lane]   = qx * qx + qy * qy + qz * qz - R2;
        qc[lane][0] = qx; qc[lane][1] = qy; qc[lane][2] = qz;
    }
    __syncthreads();

    // A matrix 16x4 (f32): VGPR0 = K0(lanes 0-15)/K2(16-31), VGPR1 = K1/K3
    // A row = (qx, qy, qz, 1)
    v2f amat;
    amat.x = (lane < 16) ? qx : qz;
    amat.y = (lane < 16) ? qy : 1.0f;

    // C matrix 16x16 (f32): VGPR v holds row v (lanes 0-15) / row v+8 (16-31)
    const int rofs = (lane >> 4) << 3;
    v8f cmat;
#pragma unroll
    for (int v = 0; v < 8; ++v) cmat[v] = qn2[v + rofs];

    // Per-row sample counts live in registers: creg[v] == count for row v+rofs.
    // The ballot half-mask is uniform across a half-wave, so all 16 lanes of a
    // half maintain identical copies — no LDS traffic in the scan loop.
    int creg[8];
#pragma unroll
    for (int v = 0; v < 8; ++v) creg[v] = 0;

    const float* xb = xyz + (size_t)b * NPTS * 3;

    // ---- scan all points, 16 per WMMA ----
    for (int n0 = 0; n0 < NPTS; n0 += 16) {
        int n = n0 + m;
        const float* pp = xb + (size_t)n * 3;
        if (n0 + 16 < NPTS) __builtin_prefetch(pp + 48, 0, 1);  // global_prefetch_b8
        float px = pp[0], py = pp[1], pz = pp[2];
        float pn2 = px * px + py * py + pz * pz;

        // B matrix 4x16 (f32): column n = (-2px, -2py, -2pz, |p|^2)
        v2f bmat;
        bmat.x = (lane < 16) ? (-2.0f * px) : (-2.0f * pz);
        bmat.y = (lane < 16) ? (-2.0f * py) : pn2;

        // D = A*B + C = d^2 - r^2  (in-ball <=> D < 0)
        v8f d = __builtin_amdgcn_wmma_f32_16x16x4_f32(
            false, amat, false, bmat, (short)0, cmat, false, false);

#pragma unroll
        for (int v = 0; v < 8; ++v) {
            bool inb = d[v] < 0.0f;
            unsigned mask = (unsigned)__ballot(inb);  // wave32: low 32 bits
            unsigned half = (lane < 16) ? (mask & 0xFFFFu) : (mask >> 16);
            int basec = creg[v];
            int pos = basec + __popc(half & ((1u << m) - 1u));
            if (inb && pos < NSAMP) ibuf[v + rofs][pos] = n;  // ascending order
            int nc = basec + __popc(half);
            creg[v] = nc > NSAMP ? NSAMP : nc;
        }

        // uniform early exit once every row is full (typical ball >> 64 points)
        int mn = creg[0];
#pragma unroll
        for (int v = 1; v < 8; ++v) mn = mn < creg[v] ? mn : creg[v];
        if ((unsigned)__ballot(mn >= NSAMP) == 0xFFFFFFFFu) break;
    }

    // publish per-row counts for the fill/gather phase
    if (m == 0) {
#pragma unroll
        for (int v = 0; v < 8; ++v) cnt[v + rofs] = creg[v];
    }
    __syncthreads();

    // ---- fill unfilled slots with first hit (0 if no hits) ----
    for (int p = lane; p < 16 * NSAMP; p += 32) {
        int row = p >> 6, slot = p & 63;
        int c = cnt[row];
        int first = (c > 0) ? ibuf[row][0] : 0;
        if (slot >= c) ibuf[row][slot] = first;
    }
    __syncthreads();

    // ---- fused gather + subtract + concat, coalesced writes ----
    const float* pb = points + (size_t)b * NPTS * NCH;
    float* ob = out + ((size_t)b * NQ + m0) * NSAMP * OCH;
#pragma unroll 4
    for (int p = 0; p < 16 * NSAMP; ++p) {
        int row = p >> 6, s = p & 63;
        int i = ibuf[row][s];                       // LDS broadcast (uniform addr)
        const float* xp = xb + (size_t)i * 3;
        const float* fp = pb + (size_t)i * NCH;
        float* op = ob + ((size_t)row * NSAMP + s) * OCH;
        if (lane < 3) op[lane] = xp[lane] - qc[row][lane];
        op[3  + lane] = fp[lane];                   // channels 0..31
        op[35 + lane] = fp[32 + lane];              // channels 32..63
    }
}

extern "C" void kernel_launch(void* const* d_in, const int* in_sizes, int n_in,
                              void* d_out, int out_size, void* d_ws, size_t ws_size,
                              hipStream_t stream) {
    const float* xyz     = (const float*)d_in[0];   // (8, 8192, 3)
    const float* new_xyz = (const float*)d_in[1];   // (8, 2048, 3)
    const float* points  = (const float*)d_in[2];   // (8, 8192, 64)
    float* out = (float*)d_out;                     // (8, 2048, 64, 67)
    (void)in_sizes; (void)n_in; (void)out_size; (void)d_ws; (void)ws_size;

    dim3 grid(NB * (NQ / 16));                      // 1024 tiles
    ballquery_group_wmma<<<grid, 32, 0, stream>>>(xyz, new_xyz, points, out);
}